// JointRegAndParc_26388279067311
// MI455X (gfx1250) — compile-verified
//
#include <hip/hip_runtime.h>

// ---------------------------------------------------------------------------
// JointRegAndParc (icosahedral 2-stream UNet) for MI455X / gfx1250.
//
// All dense math runs through V_WMMA_F32_16X16X4_F32 (fp32 WMMA): the network
// is gather-bandwidth-bound (~2 GB of one-ring gathers vs ~18 GFLOP at
// 23.3 TB/s HBM), so we keep full fp32 precision; WMMA still offloads the
// MACs from VALU and leaves VALU free for gather address math.
// ---------------------------------------------------------------------------

typedef float v2f __attribute__((ext_vector_type(2)));
typedef float v8f __attribute__((ext_vector_type(8)));

// ---------------------------------------------------------------------------
// Fused (optional gather) GEMM:
//   out[r][c] = sum_k A(r,k) * W[k][c] + bias[c],   r < M, c < cout
//   A(r,k)    = X[ idx(r, k/cin)*cin + (k%cin) ]
//   idx(r,j)  = ng ? ng[r*7 + j] : r     (nn = 7 one-ring conv, nn = 1 GEMM)
//
// One wave (32 lanes) owns a 16-row x 32-col tile; K consumed 4 per
// v_wmma_f32_16x16x4_f32.  Fragment layouts per CDNA5 ISA 7.12.2:
//   A 16x4 : lane l -> row l&15, K = kbase + 2*(l>>4) + elem
//   B 4x16 : lane l -> col l&15, K = kbase + 2*(l>>4) + elem
//   C/D    : vgpr g, lane l -> row g + 8*(l>>4), col l&15
//
// Hot loop is branch-free: column indices are clamped (invalid columns
// accumulate garbage that is never stored), A rows are clamped in s_nb,
// and the only K-tail case (cin % 4 != 0, i.e. the first conv with cin=2)
// takes a separate uniform slow path with cndmask-zeroed operands.
// ---------------------------------------------------------------------------
__global__ __launch_bounds__(32) void gemm_gather_wmma(
    const float* __restrict__ X, const int* __restrict__ ng,
    const float* __restrict__ W, const float* __restrict__ bias,
    float* __restrict__ out, int M, int cin, int cout, int nn)
{
    __shared__ int s_nb[112];                 // 16 rows x up to 7 neighbor ids
    const int lane = threadIdx.x;
    const int tr = blockIdx.x, ct = blockIdx.y;
    const int rl = lane & 15;                 // A-row / B-col within tile
    const int half = lane >> 4;               // K-pair selector

    if (ng) {
        for (int t = lane; t < 112; t += 32) {
            int rr = tr * 16 + t / 7;
            if (rr >= M) rr = M - 1;          // clamp: masked at store
            s_nb[t] = ng[rr * 7 + t % 7];
        }
    } else if (lane < 16) {
        int rr = tr * 16 + lane;
        s_nb[lane] = (rr >= M) ? M - 1 : rr;
    }
    __syncthreads();

    const int col0 = ct * 32 + rl;
    const int col1 = col0 + 16;
    const bool vc0 = col0 < cout, vc1 = col1 < cout;
    const int col0c = vc0 ? col0 : cout - 1;  // clamped address; never stored
    const int col1c = vc1 ? col1 : cout - 1;
    const size_t cstep = 4 * (size_t)cout;

    v8f acc0 = {0.f, 0.f, 0.f, 0.f, 0.f, 0.f, 0.f, 0.f};
    v8f acc1 = {0.f, 0.f, 0.f, 0.f, 0.f, 0.f, 0.f, 0.f};

    for (int j = 0; j < nn; ++j) {
        const long long nbj = s_nb[rl * nn + j];
        const float* __restrict__ Ar = X + nbj * cin;
        if (j + 1 < nn)                        // emits global_prefetch_b8
            __builtin_prefetch(X + (long long)s_nb[rl * nn + j + 1] * cin, 0, 3);
        const int kb = j * cin;

        if ((cin & 3) == 0) {
            // ---- fast path: fully unconditional, branch-free loads --------
            const float* __restrict__ ap  = Ar + 2 * half;           // 8B aligned
            const float* __restrict__ bp0 = W + (long long)(kb + 2 * half) * cout + col0c;
            const float* __restrict__ bp1 = W + (long long)(kb + 2 * half) * cout + col1c;
#pragma unroll 2
            for (int c0 = 0; c0 < cin; c0 += 4) {
                v2f a = *(const v2f*)ap;                             // load_b64
                v2f b0, b1;
                b0.x = bp0[0]; b0.y = bp0[cout];
                b1.x = bp1[0]; b1.y = bp1[cout];
                acc0 = __builtin_amdgcn_wmma_f32_16x16x4_f32(
                    false, a, false, b0, (short)0, acc0, false, false);
                acc1 = __builtin_amdgcn_wmma_f32_16x16x4_f32(
                    false, a, false, b1, (short)0, acc1, false, false);
                ap += 4; bp0 += cstep; bp1 += cstep;
            }
        } else {
            // ---- remainder-safe path (first conv only: cin == 2) ----------
            for (int c0 = 0; c0 < cin; c0 += 4) {
                const int ca = c0 + 2 * half;
                const bool va0 = ca < cin, va1 = (ca + 1) < cin;
                const int ca0 = va0 ? ca : 0;        // clamped safe addresses
                const int ca1 = va1 ? ca + 1 : 0;
                const float ax = Ar[ca0], ay = Ar[ca1];
                const long long k0 = kb + ca0, k1 = kb + ca1;
                const float t00 = W[k0 * cout + col0c], t01 = W[k1 * cout + col0c];
                const float t10 = W[k0 * cout + col1c], t11 = W[k1 * cout + col1c];
                v2f a, b0, b1;                        // zero both: no NaN leak
                a.x  = va0 ? ax  : 0.f;  a.y  = va1 ? ay  : 0.f;
                b0.x = va0 ? t00 : 0.f;  b0.y = va1 ? t01 : 0.f;
                b1.x = va0 ? t10 : 0.f;  b1.y = va1 ? t11 : 0.f;
                acc0 = __builtin_amdgcn_wmma_f32_16x16x4_f32(
                    false, a, false, b0, (short)0, acc0, false, false);
                acc1 = __builtin_amdgcn_wmma_f32_16x16x4_f32(
                    false, a, false, b1, (short)0, acc1, false, false);
            }
        }
    }

    const float bi0 = (bias && vc0) ? bias[col0] : 0.0f;
    const float bi1 = (bias && vc1) ? bias[col1] : 0.0f;
#pragma unroll
    for (int g = 0; g < 8; ++g) {
        const int rg = tr * 16 + g + 8 * half;
        if (rg < M) {
            if (vc0) out[(long long)rg * cout + col0] = acc0[g] + bi0;
            if (vc1) out[(long long)rg * cout + col1] = acc1[g] + bi1;
        }
    }
}

// Per-channel batch stats: mv[c] = mean, mv[C+c] = population variance.
__global__ void bn_stats(const float* __restrict__ y, float* __restrict__ mv,
                         int M, int C)
{
    const int c = blockIdx.x;
    const int tid = threadIdx.x;
    float s = 0.f, q = 0.f;
    for (int r = tid; r < M; r += 256) {
        float v = y[(long long)r * C + c];
        s += v; q += v * v;
    }
    __shared__ float ss[256], sq[256];
    ss[tid] = s; sq[tid] = q;
    __syncthreads();
    for (int w = 128; w > 0; w >>= 1) {
        if (tid < w) { ss[tid] += ss[tid + w]; sq[tid] += sq[tid + w]; }
        __syncthreads();
    }
    if (tid == 0) {
        float m = ss[0] / (float)M;
        float v = sq[0] / (float)M - m * m;
        mv[c] = m;
        mv[C + c] = v > 0.f ? v : 0.f;
    }
}

// In-place BN (batch stats) + LeakyReLU(0.2).
__global__ void bn_apply_leaky(float* __restrict__ y, const float* __restrict__ mv,
                               const float* __restrict__ g, const float* __restrict__ be,
                               long long total, int C)
{
    long long i = (long long)blockIdx.x * blockDim.x + threadIdx.x;
    if (i >= total) return;
    int c = (int)(i % C);
    float m = mv[c], v = mv[C + c];
    float t = (y[i] - m) * (g[c] * rsqrtf(v + 1e-5f)) + be[c];
    y[i] = t >= 0.f ? t : 0.2f * t;
}

// Mean-pool of 7 gathered rows: out[v][c] = mean_j X[pno[v*7+j]][c].
__global__ void pool7(const float* __restrict__ X, const int* __restrict__ pno,
                      float* __restrict__ out, int M, int C)
{
    long long i = (long long)blockIdx.x * blockDim.x + threadIdx.x;
    if (i >= (long long)M * C) return;
    int v = (int)(i / C);
    int c = (int)(i - (long long)v * C);
    float s = 0.f;
#pragma unroll
    for (int j = 0; j < 7; ++j) s += X[(long long)pno[v * 7 + j] * C + c];
    out[i] = s * (1.0f / 7.0f);
}

// Up-conv recombine. t is [Nr*7, C] (flat view of [Nr, 7C]).
//   v <  Nr : out[v][c] = t[top[v]][c]
//   v >= Nr : exact jnp reshape(-1,C,2).mean(-1) interleaving of t[down].
__global__ void upcombine(const float* __restrict__ t, const int* __restrict__ top,
                          const int* __restrict__ down, float* __restrict__ out,
                          int Nr, int Nn, int C)
{
    long long i = (long long)blockIdx.x * blockDim.x + threadIdx.x;
    if (i >= (long long)Nn * C) return;
    int v = (int)(i / C);
    int c = (int)(i - (long long)v * C);
    float r;
    if (v < Nr) {
        r = t[(long long)top[v] * C + c];
    } else {
        int m = v - Nr;
        long long f0 = (long long)(2 * m) * C + 2 * c;
        long long f1 = f0 + 1;
        int r0 = (int)(f0 / C); int c0 = (int)(f0 - (long long)r0 * C);
        int r1 = (int)(f1 / C); int c1 = (int)(f1 - (long long)r1 * C);
        r = 0.5f * (t[(long long)down[r0] * C + c0] +
                    t[(long long)down[r1] * C + c1]);
    }
    out[i] = r;
}

// Copy src [M,Cs] into columns [off, off+Cs) of dst [M,Cd] (for concats).
__global__ void copy_cols(const float* __restrict__ src, float* __restrict__ dst,
                          int M, int Cs, int Cd, int off)
{
    long long i = (long long)blockIdx.x * blockDim.x + threadIdx.x;
    if (i >= (long long)M * Cs) return;
    int r = (int)(i / Cs);
    int c = (int)(i - (long long)r * Cs);
    dst[(long long)r * Cd + off + c] = src[i];
}

// ---------------------------------------------------------------------------
// Host orchestration
// ---------------------------------------------------------------------------
struct Blk { const float *W1, *b1, *g1, *be1, *W2, *b2, *g2, *be2, *uW, *ub; };

static inline unsigned eb(long long n) { return (unsigned)((n + 255) / 256); }

extern "C" void kernel_launch(void* const* d_in, const int* in_sizes, int n_in,
                              void* d_out, int out_size, void* d_ws, size_t ws_size,
                              hipStream_t stream)
{
    (void)in_sizes; (void)n_in; (void)out_size; (void)ws_size;

    static const int NSv[5]  = {163842, 40962, 10242, 2562, 642};
    static const int CHSv[6] = {2, 16, 32, 64, 128, 256};

    // ---- inputs: dict insertion order at every nesting level --------------
    int p = 0;
    auto nextF = [&]() -> const float* { return (const float*)d_in[p++]; };
    auto nextI = [&]() -> const int*   { return (const int*)d_in[p++]; };

    const float* x1 = nextF();
    const float* x2 = nextF();
    Blk down[5], upp[4], upr[4];
    for (int i = 0; i < 5; ++i) {
        down[i].W1 = nextF(); down[i].b1 = nextF(); down[i].g1 = nextF(); down[i].be1 = nextF();
        down[i].W2 = nextF(); down[i].b2 = nextF(); down[i].g2 = nextF(); down[i].be2 = nextF();
        down[i].uW = nullptr; down[i].ub = nullptr;
    }
    for (int i = 0; i < 4; ++i) {
        upp[i].W1 = nextF(); upp[i].b1 = nextF(); upp[i].g1 = nextF(); upp[i].be1 = nextF();
        upp[i].W2 = nextF(); upp[i].b2 = nextF(); upp[i].g2 = nextF(); upp[i].be2 = nextF();
        upp[i].uW = nextF(); upp[i].ub = nextF();
    }
    for (int i = 0; i < 4; ++i) {
        upr[i].W1 = nextF(); upr[i].b1 = nextF(); upr[i].g1 = nextF(); upr[i].be1 = nextF();
        upr[i].W2 = nextF(); upr[i].b2 = nextF(); upr[i].g2 = nextF(); upr[i].be2 = nextF();
        upr[i].uW = nextF(); upr[i].ub = nextF();
    }
    const float* parcW = nextF(); const float* parcb = nextF();
    const float* regW  = nextF(); const float* regb  = nextF();
    const int* neigh[5]; for (int i = 0; i < 5; ++i) neigh[i] = nextI();
    const int* utop[4];  for (int i = 0; i < 4; ++i) utop[i]  = nextI();
    const int* udown[4]; for (int i = 0; i < 4; ++i) udown[i] = nextI();

    // ---- workspace layout (floats) ----------------------------------------
    float* w = (float*)d_ws;
    size_t off = 0;
    auto alloc = [&](size_t n) { float* r = w + off; off += n; return r; };
    float *XS1[5], *XS2[5], *YS[5];             // xs[l+1]: NS[l] rows, CHS[l+1] ch
    for (int l = 0; l < 5; ++l) {
        size_t n = (size_t)NSv[l] * CHSv[l + 1];
        XS1[l] = alloc(n); XS2[l] = alloc(n); YS[l] = alloc(2 * n);
    }
    float* PT = alloc((size_t)40962 * 16);      // pooled input
    float* T1 = alloc((size_t)163842 * 16);     // conv1 pre-BN
    float* TU = alloc((size_t)40962 * 112);     // upconv GEMM out [Nr, 7*cout]
    float* UC = alloc((size_t)163842 * 16);     // recombined upconv
    float* XC = alloc((size_t)163842 * 48);     // concat input of up conv1
    float* HA = alloc((size_t)163842 * 16);     // up-path ping
    float* HB = alloc((size_t)163842 * 16);     // up-path pong
    float* MV = alloc(1024);                    // bn mean/var (<=512 ch)

    hipStream_t st = stream;
    auto gemm = [&](const float* Xp, const int* ngp, const float* Wp, const float* bp,
                    float* op, int M, int cin, int cout, int nn) {
        dim3 g((M + 15) / 16, (cout + 31) / 32);
        gemm_gather_wmma<<<g, dim3(32), 0, st>>>(Xp, ngp, Wp, bp, op, M, cin, cout, nn);
    };
    auto bn = [&](float* y, const float* gg, const float* bb, int M, int C) {
        bn_stats<<<C, 256, 0, st>>>(y, MV, M, C);
        long long tot = (long long)M * C;
        bn_apply_leaky<<<eb(tot), 256, 0, st>>>(y, MV, gg, bb, tot, C);
    };

    // ---- UNet down path ----------------------------------------------------
    auto run_down = [&](const float* x0, float* const* XS) {
        const float* cur = x0;
        for (int i = 0; i < 5; ++i) {
            int M = NSv[i], cin = CHSv[i], cout = CHSv[i + 1];
            const float* src = cur;
            if (i > 0) {
                long long tot = (long long)M * cin;
                pool7<<<eb(tot), 256, 0, st>>>(cur, neigh[i - 1], PT, M, cin);
                src = PT;
            }
            gemm(src, neigh[i], down[i].W1, down[i].b1, T1, M, cin, cout, 7);
            bn(T1, down[i].g1, down[i].be1, M, cout);
            gemm(T1, neigh[i], down[i].W2, down[i].b2, XS[i], M, cout, cout, 7);
            bn(XS[i], down[i].g2, down[i].be2, M, cout);
            cur = XS[i];
        }
    };

    // ---- UNet up path (parc: mul=1 with XS skips; reg: mul=2 with YS) -----
    auto run_up = [&](const float* h0, int ch0, const Blk* B,
                      float* const* SK, int mul) -> const float* {
        const float* h = h0; int ch = ch0;
        for (int s = 0; s < 4; ++s) {
            int j = 3 - s;
            int Nr = NSv[j + 1], Nn = NSv[j];
            int cout = CHSv[4 - s];
            gemm(h, nullptr, B[s].uW, B[s].ub, TU, Nr, ch, 7 * cout, 1);
            long long totu = (long long)Nn * cout;
            upcombine<<<eb(totu), 256, 0, st>>>(TU, utop[j], udown[j], UC, Nr, Nn, cout);
            int cs = mul * CHSv[4 - s];
            int joint = cout + cs;
            copy_cols<<<eb((long long)Nn * cout), 256, 0, st>>>(UC, XC, Nn, cout, joint, 0);
            copy_cols<<<eb((long long)Nn * cs), 256, 0, st>>>(SK[3 - s], XC, Nn, cs, joint, cout);
            gemm(XC, neigh[j], B[s].W1, B[s].b1, T1, Nn, joint, cout, 7);
            bn(T1, B[s].g1, B[s].be1, Nn, cout);
            float* hn = (s & 1) ? HB : HA;
            gemm(T1, neigh[j], B[s].W2, B[s].b2, hn, Nn, cout, cout, 7);
            bn(hn, B[s].g2, B[s].be2, Nn, cout);
            h = hn; ch = cout;
        }
        return h;
    };

    float* out = (float*)d_out;

    // stream 1 (parc)
    run_down(x1, XS1);
    const float* h1 = run_up(XS1[4], CHSv[5], upp, XS1, 1);
    gemm(h1, nullptr, parcW, parcb, out, NSv[0], 16, 36, 1);

    // stream 2 (parc)
    run_down(x2, XS2);
    const float* h2 = run_up(XS2[4], CHSv[5], upp, XS2, 1);
    gemm(h2, nullptr, parcW, parcb, out + (size_t)NSv[0] * 36, NSv[0], 16, 36, 1);

    // ys[l] = concat(x1s[l], x2s[l])
    for (int l = 0; l < 5; ++l) {
        int rows = NSv[l], C = CHSv[l + 1];
        copy_cols<<<eb((long long)rows * C), 256, 0, st>>>(XS1[l], YS[l], rows, C, 2 * C, 0);
        copy_cols<<<eb((long long)rows * C), 256, 0, st>>>(XS2[l], YS[l], rows, C, 2 * C, C);
    }

    // registration branch
    const float* hr = run_up(YS[4], 2 * CHSv[5], upr, YS, 2);
    gemm(hr, nullptr, regW, regb, out + (size_t)NSv[0] * 72, NSv[0], 16, 2, 1);
}